// GCNModel_88914412962545
// MI455X (gfx1250) — compile-verified
//
#include <hip/hip_runtime.h>

// MI455X / gfx1250 GCN edge-prediction pipeline.
// fp32 WMMA (V_WMMA_F32_16X16X4_F32) for all GEMMs; L2-resident scatter-add;
// edge head folded into per-node scalars so the E x 27 "ef" tensor never
// touches memory. This revision batches A-fragment loads ahead of the WMMA
// chains (clause + single wait instead of load/wait/wmma lockstep) and uses
// unguarded pointer-increment stores on full tiles.

typedef __attribute__((ext_vector_type(2))) float v2f;
typedef __attribute__((ext_vector_type(4))) float v4f;
typedef __attribute__((ext_vector_type(8))) float v8f;

static __device__ __forceinline__ v8f wmma_f32_k4(v2f a, v2f b, v8f c) {
  // D(16x16,f32) = A(16x4,f32) * B(4x16,f32) + C
  return __builtin_amdgcn_wmma_f32_16x16x4_f32(
      /*neg_a=*/false, a, /*neg_b=*/false, b,
      /*c_mod=*/(short)0, c, /*reuse_a=*/false, /*reuse_b=*/false);
}

static __device__ __forceinline__ void wave_lds_sync() {
  // wave-local LDS producer->consumer ordering (cross-lane): drain DScnt.
  __builtin_amdgcn_wave_barrier();
  asm volatile("s_wait_dscnt 0" ::: "memory");
  __builtin_amdgcn_wave_barrier();
}

// ---------------------------------------------------------------- utilities
__global__ void fill4_kernel(v4f* __restrict__ p, long long n4, float v) {
  long long i = blockIdx.x * (long long)blockDim.x + threadIdx.x;
  if (i < n4) p[i] = (v4f){v, v, v, v};
}

__global__ void deg_init_kernel(float* __restrict__ deg, int n) {
  int i = blockIdx.x * blockDim.x + threadIdx.x;
  if (i < n) deg[i] = 1.0f;  // self-loop
}

__global__ void deg_accum_kernel(const int* __restrict__ eidx,
                                 float* __restrict__ deg, int E_) {
  int e = blockIdx.x * blockDim.x + threadIdx.x;
  if (e < E_) atomicAdd(&deg[eidx[E_ + e]], 1.0f);  // target-node degree
}

__global__ void dinv_kernel(const float* __restrict__ deg,
                            float* __restrict__ dinv, int n) {
  int i = blockIdx.x * blockDim.x + threadIdx.x;
  if (i < n) dinv[i] = rsqrtf(deg[i]);  // deg >= 1 always
}

// ------------------------------------------------- node GEMM: [N,32]@[K,27]
// out is [N,32] m-major (cols 27..31 are garbage here; consumers mask them).
__global__ __launch_bounds__(256) void node_gemm_kernel(
    const float* __restrict__ in, const float* __restrict__ w,
    float* __restrict__ out, int nRows, int kvalid) {
  const int lane = threadIdx.x & 31;
  const int mrow = lane & 15;
  const int kh   = lane >> 4;
  const int wavesPerBlock = blockDim.x >> 5;
  int wave   = blockIdx.x * wavesPerBlock + (threadIdx.x >> 5);
  int nWaves = gridDim.x * wavesPerBlock;
  int nTiles = (nRows + 15) >> 4;

  // B fragments: 2 n-tiles x 8 k-steps, resident in VGPRs.
  v2f bf[2][8];
#pragma unroll
  for (int nt = 0; nt < 2; ++nt) {
    int n = nt * 16 + mrow;
#pragma unroll
    for (int s = 0; s < 8; ++s)
#pragma unroll
      for (int v = 0; v < 2; ++v) {
        int k = 4 * s + 2 * kh + v;
        bf[nt][s][v] = (k < kvalid && n < 27) ? w[k * 27 + n] : 0.0f;
      }
  }

  for (int t = wave; t < nTiles; t += nWaves) {
    int row = t * 16 + mrow;
    if (row >= nRows) row = nRows - 1;  // clamp loads; stores guarded below
    const float* arow = in + (size_t)row * 32 + 2 * kh;

    // Batch all A fragments: one load clause + one wait, then 16 WMMAs.
    v2f a[8];
#pragma unroll
    for (int s = 0; s < 8; ++s) a[s] = *(const v2f*)(arow + 4 * s);

    v8f c0 = {}, c1 = {};
#pragma unroll
    for (int s = 0; s < 8; ++s) {
      c0 = wmma_f32_k4(a[s], bf[0][s], c0);
      c1 = wmma_f32_k4(a[s], bf[1][s], c1);
    }

    // D layout: vgpr v, lanes 0-15 -> row v, lanes 16-31 -> row v+8.
    int nIdx = lane & 15;
    int mBase = (lane >> 4) * 8;
    float* obase = out + ((size_t)(t * 16 + mBase)) * 32 + nIdx;
    if (t * 16 + 16 <= nRows) {  // full tile: unguarded streaming stores
#pragma unroll
      for (int v = 0; v < 8; ++v) {
        obase[0]  = c0[v];
        obase[16] = c1[v];
        obase += 32;
      }
    } else {
      int orow = t * 16 + mBase;
#pragma unroll
      for (int v = 0; v < 8; ++v) {
        if (orow + v < nRows) {
          obase[0]  = c0[v];
          obase[16] = c1[v];
        }
        obase += 32;
      }
    }
  }
}

// --------------------------------------------- scatter-add over edges (L2)
__global__ void scatter_kernel(const float* __restrict__ t,
                               const int* __restrict__ eidx,
                               const float* __restrict__ dinv,
                               float* __restrict__ agg, int E_) {
  int e = blockIdx.x * blockDim.x + threadIdx.x;
  if (e >= E_) return;
  int r = eidx[e], c = eidx[E_ + e];
  float s = dinv[r] * dinv[c];
  const float* src = t + (size_t)r * 32;
  float* dst = agg + (size_t)c * 32;
#pragma unroll
  for (int f = 0; f < 27; ++f) atomicAdd(dst + f, s * src[f]);
}

// h1 = relu(agg + dinv^2 * t + b1); pad cols 27..31 with zeros for GEMM2.
__global__ void finalize1_kernel(const float* __restrict__ t,
                                 const float* __restrict__ agg,
                                 const float* __restrict__ dinv,
                                 const float* __restrict__ b1,
                                 float* __restrict__ h1, int n) {
  long long idx = blockIdx.x * (long long)blockDim.x + threadIdx.x;
  if (idx >= (long long)n * 32) return;
  int node = (int)(idx >> 5), f = (int)(idx & 31);
  float v = 0.0f;
  if (f < 27) {
    float d = dinv[node];
    v = fmaxf(agg[idx] + d * d * t[idx] + b1[f], 0.0f);
  }
  h1[idx] = v;
}

// h = agg + dinv^2*t + b2 ; fold head: hr = h.fcw[0:27], hc = h.fcw[27:54]
__global__ void finalize2_kernel(const float* __restrict__ t,
                                 const float* __restrict__ agg,
                                 const float* __restrict__ dinv,
                                 const float* __restrict__ b2,
                                 const float* __restrict__ fcw,
                                 float* __restrict__ hr, float* __restrict__ hc,
                                 int n) {
  int i = blockIdx.x * blockDim.x + threadIdx.x;
  if (i >= n) return;
  float d = dinv[i];
  float d2 = d * d;
  const float* ti = t + (size_t)i * 32;
  const float* ai = agg + (size_t)i * 32;
  float sr = 0.0f, sc = 0.0f;
#pragma unroll
  for (int f = 0; f < 27; ++f) {
    float hv = ai[f] + d2 * ti[f] + b2[f];
    sr += hv * fcw[f];
    sc += hv * fcw[27 + f];
  }
  hr[i] = sr;
  hc[i] = sc;
}

// ---------------------------------- edge MLP + final head, fused, 7.4 GFLOP
// per 16-edge tile: 8 WMMA (16->27) + 16 WMMA (27->27), ef stays on-WGP.
__global__ __launch_bounds__(256) void edge_kernel(
    const float* __restrict__ ea,   // [E,16]
    const int* __restrict__ eidx,   // [2,E]
    const float* __restrict__ ew1, const float* __restrict__ eb1,
    const float* __restrict__ ew2, const float* __restrict__ eb2,
    const float* __restrict__ fcw, const float* __restrict__ fcb,
    const float* __restrict__ hr, const float* __restrict__ hc,
    float* __restrict__ outp, int E_) {
  __shared__ float tileLds[8][16 * 32];  // 2KB per wave staging
  __shared__ float fcw3s[28];
  if (threadIdx.x < 27) fcw3s[threadIdx.x] = fcw[54 + threadIdx.x];
  if (threadIdx.x == 27) fcw3s[27] = fcb[0];
  __syncthreads();

  const int lane = threadIdx.x & 31;
  const int mrow = lane & 15;
  const int kh   = lane >> 4;
  float* wl = tileLds[threadIdx.x >> 5];

  // Resident B fragments + biases.
  v2f b1f[2][4], b2f[2][8];
  float eb1v[2], eb2v[2];
#pragma unroll
  for (int nt = 0; nt < 2; ++nt) {
    int n = nt * 16 + mrow;
    bool nok = (n < 27);
    eb1v[nt] = nok ? eb1[n] : 0.0f;
    eb2v[nt] = nok ? eb2[n] : 0.0f;
#pragma unroll
    for (int s = 0; s < 4; ++s)
#pragma unroll
      for (int v = 0; v < 2; ++v) {
        int k = 4 * s + 2 * kh + v;  // < 16
        b1f[nt][s][v] = nok ? ew1[k * 27 + n] : 0.0f;
      }
#pragma unroll
    for (int s = 0; s < 8; ++s)
#pragma unroll
      for (int v = 0; v < 2; ++v) {
        int k = 4 * s + 2 * kh + v;  // < 32
        b2f[nt][s][v] = (nok && k < 27) ? ew2[k * 27 + n] : 0.0f;
      }
  }

  int nTiles = (E_ + 15) >> 4;
  int wavesPerBlock = blockDim.x >> 5;
  int wave   = blockIdx.x * wavesPerBlock + (threadIdx.x >> 5);
  int nWaves = gridDim.x * wavesPerBlock;

  for (int t = wave; t < nTiles; t += nWaves) {
    int e0 = t * 16;
    int erow = e0 + mrow;
    if (erow >= E_) erow = E_ - 1;
    const float* arow = ea + (size_t)erow * 16 + 2 * kh;

    // Prefetch next tile's edge_attr row for this wave (HBM stream).
    {
      int enext = e0 + nWaves * 16 + mrow;
      if (enext < E_) __builtin_prefetch(ea + (size_t)enext * 16, 0, 1);
    }

    // GEMM1: [16,16] @ [16,27] -- batch loads, then WMMA chain.
    v2f a1[4];
#pragma unroll
    for (int s = 0; s < 4; ++s) a1[s] = *(const v2f*)(arow + 4 * s);
    v8f c0 = {}, c1 = {};
#pragma unroll
    for (int s = 0; s < 4; ++s) {
      c0 = wmma_f32_k4(a1[s], b1f[0][s], c0);
      c1 = wmma_f32_k4(a1[s], b1f[1][s], c1);
    }

    // bias + relu -> LDS in A-friendly (m-major) layout
    {
      int nIdx = lane & 15, mBase = (lane >> 4) * 8;
      float* wp = wl + mBase * 32 + nIdx;
#pragma unroll
      for (int v = 0; v < 8; ++v) {
        wp[0]  = fmaxf(c0[v] + eb1v[0], 0.0f);
        wp[16] = fmaxf(c1[v] + eb1v[1], 0.0f);
        wp += 32;
      }
    }
    wave_lds_sync();

    // GEMM2: [16,32pad] @ [27,27] -- batch LDS loads, then WMMA chain.
    v2f a2[8];
    {
      const float* lrow = wl + mrow * 32 + 2 * kh;
#pragma unroll
      for (int s = 0; s < 8; ++s) a2[s] = *(const v2f*)(lrow + 4 * s);
    }
    v8f d0 = {}, d1 = {};
#pragma unroll
    for (int s = 0; s < 8; ++s) {
      d0 = wmma_f32_k4(a2[s], b2f[0][s], d0);
      d1 = wmma_f32_k4(a2[s], b2f[1][s], d1);
    }
    wave_lds_sync();

    // ef + bias -> LDS (per-edge rows)
    {
      int nIdx = lane & 15, mBase = (lane >> 4) * 8;
      float* wp = wl + mBase * 32 + nIdx;
#pragma unroll
      for (int v = 0; v < 8; ++v) {
        wp[0]  = d0[v] + eb2v[0];
        wp[16] = d1[v] + eb2v[1];
        wp += 32;
      }
    }
    wave_lds_sync();

    // final head: pred = ef.fcw3 + fcb + hr[row] + hc[col]
    if (lane < 16) {
      int e = e0 + lane;
      if (e < E_) {
        int r = eidx[e], c = eidx[E_ + e];
        float tail = hr[r] + hc[c] + fcw3s[27];
        float sum = 0.0f;
        const float* wrow = wl + lane * 32;
#pragma unroll
        for (int f = 0; f < 27; ++f) sum += wrow[f] * fcw3s[f];
        outp[e] = sum + tail;
      }
    }
    wave_lds_sync();
  }
}

// ------------------------------------------------------------------ driver
extern "C" void kernel_launch(void* const* d_in, const int* in_sizes, int n_in,
                              void* d_out, int out_size, void* d_ws,
                              size_t ws_size, hipStream_t stream) {
  const float* x   = (const float*)d_in[0];
  const int*   eix = (const int*)d_in[1];
  const float* ea  = (const float*)d_in[2];
  const float* W1  = (const float*)d_in[3];
  const float* b1  = (const float*)d_in[4];
  const float* W2  = (const float*)d_in[5];
  const float* b2  = (const float*)d_in[6];
  const float* ew1 = (const float*)d_in[7];
  const float* eb1 = (const float*)d_in[8];
  const float* ew2 = (const float*)d_in[9];
  const float* eb2 = (const float*)d_in[10];
  const float* fcw = (const float*)d_in[11];
  const float* fcb = (const float*)d_in[12];
  float* outp = (float*)d_out;

  const int N_ = in_sizes[0] / 32;  // 100000
  const int E_ = in_sizes[1] / 2;   // 3200000

  // workspace layout (floats): all node arrays fit in 192MB L2
  float* ws   = (float*)d_ws;
  float* deg  = ws;                         // N
  float* dinv = ws + (size_t)N_;            // N
  float* hr   = ws + 2 * (size_t)N_;        // N
  float* hc   = ws + 3 * (size_t)N_;        // N
  float* t    = ws + 4 * (size_t)N_;        // N*32 (t1 then t2)
  float* agg  = t + 32 * (size_t)N_;        // N*32
  float* h1   = agg + 32 * (size_t)N_;      // N*32

  dim3 blk(256);
  int nodeBlocks  = (N_ + 255) / 256;
  long long n4    = ((long long)N_ * 32) / 4;
  int fillBlocks  = (int)((n4 + 255) / 256);
  int elemBlocks  = (int)(((long long)N_ * 32 + 255) / 256);
  int edgeBlocks  = (E_ + 255) / 256;
  int gemmBlocks  = ((N_ + 15) / 16 + 7) / 8;  // 8 waves per block
  int egBlocks    = ((E_ + 15) / 16 + 7) / 8;

  // degrees / symmetric norm
  deg_init_kernel<<<nodeBlocks, blk, 0, stream>>>(deg, N_);
  deg_accum_kernel<<<edgeBlocks, blk, 0, stream>>>(eix, deg, E_);
  dinv_kernel<<<nodeBlocks, blk, 0, stream>>>(deg, dinv, N_);

  // layer 1
  node_gemm_kernel<<<gemmBlocks, blk, 0, stream>>>(x, W1, t, N_, 32);
  fill4_kernel<<<fillBlocks, blk, 0, stream>>>((v4f*)agg, n4, 0.0f);
  scatter_kernel<<<edgeBlocks, blk, 0, stream>>>(t, eix, dinv, agg, E_);
  finalize1_kernel<<<elemBlocks, blk, 0, stream>>>(t, agg, dinv, b1, h1, N_);

  // layer 2 (+ head folding into hr/hc)
  node_gemm_kernel<<<gemmBlocks, blk, 0, stream>>>(h1, W2, t, N_, 27);
  fill4_kernel<<<fillBlocks, blk, 0, stream>>>((v4f*)agg, n4, 0.0f);
  scatter_kernel<<<edgeBlocks, blk, 0, stream>>>(t, eix, dinv, agg, E_);
  finalize2_kernel<<<nodeBlocks, blk, 0, stream>>>(t, agg, dinv, b2, fcw, hr,
                                                   hc, N_);

  // fused edge MLP + prediction
  edge_kernel<<<egBlocks, blk, 0, stream>>>(ea, eix, ew1, eb1, ew2, eb2, fcw,
                                            fcb, hr, hc, outp, E_);
}